// PanopticSegmentationGenerator_15857019256887
// MI455X (gfx1250) — compile-verified
//
#include <hip/hip_runtime.h>
#include <math.h>
#include <stdint.h>

// Problem constants (from reference)
#define OUT_HW 640
#define IN_HW  160
#define NCH    20
#define NDET   100
#define MSZ    28

struct Det {
  int   ymin, xmin, ymax, xmax;  // ymax/xmax already +1 and clipped to [0,640]
  float bh, bw;                  // max(ymax-ymin,1) as float (matches ref arithmetic)
  float cls;                     // class label (float in ref)
  int   inst;                    // original index + 1
  int   valid;                   // score > 0.05 (sorted => monotone prefix)
  int   pad0, pad1, pad2;        // 48B struct
};

// -------- Kernel A: per-batch sort + detection param prep + mask binarize ----
__global__ void panoptic_prep(const float* __restrict__ scores,
                              const float* __restrict__ classes,
                              const int*   __restrict__ boxes,
                              const float* __restrict__ masks,
                              Det*      __restrict__ dets,
                              unsigned* __restrict__ bits) {
  const int b = blockIdx.x;
  const int t = threadIdx.x;
  __shared__ float s_sc[NDET];
  __shared__ int   s_order[NDET];

  if (t < NDET) s_sc[t] = scores[b * NDET + t];
  __syncthreads();

  // Stable descending rank sort (matches jnp.argsort(-scores), stable kind)
  if (t < NDET) {
    const float si = s_sc[t];
    int rank = 0;
    for (int j = 0; j < NDET; ++j) {
      const float sj = s_sc[j];
      rank += (sj > si) || (sj == si && j < t);
    }
    s_order[rank] = t;
  }
  __syncthreads();

  if (t < NDET) {
    const int  w  = s_order[t];
    const int* bx = boxes + (b * NDET + w) * 4;
    Det d;
    d.ymin = bx[0];
    d.xmin = bx[1];
    int ymax = bx[2] + 1; ymax = ymax < 0 ? 0 : (ymax > OUT_HW ? OUT_HW : ymax);
    int xmax = bx[3] + 1; xmax = xmax < 0 ? 0 : (xmax > OUT_HW ? OUT_HW : xmax);
    d.ymax = ymax; d.xmax = xmax;
    int bh = ymax - d.ymin; if (bh < 1) bh = 1;
    int bw = xmax - d.xmin; if (bw < 1) bw = 1;
    d.bh = (float)bh; d.bw = (float)bw;
    d.cls  = classes[b * NDET + w];
    d.inst = w + 1;
    d.valid = (s_sc[w] > 0.05f) ? 1 : 0;
    d.pad0 = d.pad1 = d.pad2 = 0;
    dets[b * NDET + t] = d;
  }
  __syncthreads();

  // Binarize masks (>0.5) bit-packed: one uint32 per 28-wide mask row,
  // already permuted into sorted order.
  for (int item = t; item < NDET * MSZ; item += blockDim.x) {
    const int k = item / MSZ;
    const int r = item - k * MSZ;
    const int w = s_order[k];
    const float* m = masks + ((size_t)(b * NDET + w) * MSZ + r) * MSZ;
    unsigned word = 0u;
#pragma unroll
    for (int c = 0; c < MSZ; ++c) word |= (m[c] > 0.5f ? 1u : 0u) << c;
    bits[(b * NDET + k) * MSZ + r] = word;
  }
}

// -------- Kernel B: per-pixel merge (16x16 tile per 256-thread block) -------
__global__ void panoptic_main(const float*    __restrict__ seg,
                              const Det*      __restrict__ dets,
                              const unsigned* __restrict__ bits,
                              int* __restrict__ out, int B) {
  const int b = blockIdx.z;
  const int t = threadIdx.x;

  __shared__ __attribute__((aligned(16))) unsigned s_bits[NDET * MSZ];
  __shared__ int   s_ymin[NDET], s_xmin[NDET], s_ymax[NDET], s_xmax[NDET];
  __shared__ float s_bh[NDET], s_bw[NDET], s_cls[NDET];
  __shared__ int   s_inst[NDET];
  __shared__ unsigned char s_flag[NDET];
  __shared__ short s_list[NDET];
  __shared__ int   s_m;

  // --- gfx1250 async DMA: bit-packed masks (11.2 KB) global -> LDS, B128
  //     per lane, GVS addressing (SGPR base + VGPR byte offset), tracked by
  //     ASYNCcnt. No VGPR round-trip; overlaps with the param loads below.
  {
    const unsigned ldsbase = (unsigned)(uintptr_t)(&s_bits[0]);
    const void* gsrc = (const void*)(bits + (size_t)b * NDET * MSZ);
    for (int i = t; i < (NDET * MSZ) / 4; i += blockDim.x) {  // 700 x 16B
      const unsigned lds_addr = ldsbase + (unsigned)i * 16u;
      const unsigned voff = (unsigned)i * 16u;
      asm volatile("global_load_async_to_lds_b128 %0, %1, %2"
                   :: "v"(lds_addr), "v"(voff), "s"(gsrc) : "memory");
    }
  }

  const int ty0 = blockIdx.y * 16;
  const int tx0 = blockIdx.x * 16;

  if (t < NDET) {
    Det d = dets[b * NDET + t];
    s_ymin[t] = d.ymin; s_xmin[t] = d.xmin;
    s_ymax[t] = d.ymax; s_xmax[t] = d.xmax;
    s_bh[t] = d.bh; s_bw[t] = d.bw;
    s_cls[t] = d.cls; s_inst[t] = d.inst;
    // Tile/box intersection in sorted order; box containment is an exact
    // precondition of the per-pixel inside test, and validity is a monotone
    // prefix, so compaction preserves first-hit semantics exactly.
    s_flag[t] = (unsigned char)(d.valid &&
                                d.ymin <= ty0 + 15 && d.ymax > ty0 &&
                                d.xmin <= tx0 + 15 && d.xmax > tx0);
  }

  // Drain the async LDS writes before the workgroup barrier publishes them.
  asm volatile("s_wait_asynccnt 0x0" ::: "memory");
  __syncthreads();

  if (t == 0) {
    int m = 0;
    for (int k = 0; k < NDET; ++k)
      if (s_flag[k]) s_list[m++] = (short)k;
    s_m = m;
  }
  __syncthreads();

  const int x = tx0 + (t & 15);
  const int y = ty0 + (t >> 4);

  // Bilinear tap setup (half-pixel centers, in/out scale = 0.25, edge clamp
  // == jax.image.resize 'linear' upsampling semantics).
  const float fy = ((float)y + 0.5f) * 0.25f - 0.5f;
  const float fx = ((float)x + 0.5f) * 0.25f - 0.5f;
  const int y0 = (int)floorf(fy);
  const int x0 = (int)floorf(fx);
  const float wy = fy - (float)y0;
  const float wx = fx - (float)x0;
  const int y0c = y0 < 0 ? 0 : (y0 > IN_HW - 1 ? IN_HW - 1 : y0);
  const int y1t = y0 + 1;
  const int y1c = y1t < 0 ? 0 : (y1t > IN_HW - 1 ? IN_HW - 1 : y1t);
  const int x0c = x0 < 0 ? 0 : (x0 > IN_HW - 1 ? IN_HW - 1 : x0);
  const int x1t = x0 + 1;
  const int x1c = x1t < 0 ? 0 : (x1t > IN_HW - 1 ? IN_HW - 1 : x1t);

  const float* p00 = seg + (((size_t)b * IN_HW + y0c) * IN_HW + x0c) * NCH;
  const float* p01 = seg + (((size_t)b * IN_HW + y0c) * IN_HW + x1c) * NCH;
  const float* p10 = seg + (((size_t)b * IN_HW + y1c) * IN_HW + x0c) * NCH;
  const float* p11 = seg + (((size_t)b * IN_HW + y1c) * IN_HW + x1c) * NCH;
  // global_prefetch_b8 hints: pull tap rows toward the WGP during the scan.
  __builtin_prefetch(p00, 0, 0);
  __builtin_prefetch(p10, 0, 0);

  // First-hit scan over the compacted (order-preserving) detection list.
  int cat = -1, ins = -1;
  const int m = s_m;
  for (int j = 0; j < m; ++j) {
    const int k = s_list[j];
    if (y >= s_ymin[k] && y < s_ymax[k] && x >= s_xmin[k] && x < s_xmax[k]) {
      // Match reference fp order: ((y - ymin + 0.5) * 28) / bh
      const float ty = (((float)y - (float)s_ymin[k]) + 0.5f) * 28.0f / s_bh[k];
      const float tx = (((float)x - (float)s_xmin[k]) + 0.5f) * 28.0f / s_bw[k];
      int my = (int)floorf(ty); my = my < 0 ? 0 : (my > MSZ - 1 ? MSZ - 1 : my);
      int mx = (int)floorf(tx); mx = mx < 0 ? 0 : (mx > MSZ - 1 ? MSZ - 1 : mx);
      if ((s_bits[k * MSZ + my] >> mx) & 1u) {
        cat = (int)s_cls[k];
        ins = s_inst[k];
        break;
      }
    }
  }

  // Uncovered pixels: 20-channel bilerp + first-max argmax, then stuff remap.
  if (cat < 0) {
    const float w00 = (1.0f - wy) * (1.0f - wx);
    const float w01 = (1.0f - wy) * wx;
    const float w10 = wy * (1.0f - wx);
    const float w11 = wy * wx;
    float best = -3.402823466e38f;
    int s = 0;
#pragma unroll
    for (int c = 0; c < NCH; ++c) {
      const float v = w00 * p00[c] + w01 * p01[c] + w10 * p10[c] + w11 * p11[c];
      if (v > best) { best = v; s = c; }
    }
    // s==0 -> VOID(0), s==1 -> THINGS (stays void category), s>=2 -> stuff +90
    cat = (s >= 2) ? (s + 90) : 0;
    ins = -1;
  }

  const int pix = (b * OUT_HW + y) * OUT_HW + x;
  out[pix] = cat;
  out[B * OUT_HW * OUT_HW + pix] = ins;
}

// ----------------------------------------------------------------------------
extern "C" void kernel_launch(void* const* d_in, const int* in_sizes, int n_in,
                              void* d_out, int out_size, void* d_ws, size_t ws_size,
                              hipStream_t stream) {
  const float* scores  = (const float*)d_in[0];  // (B,100)
  const float* classes = (const float*)d_in[1];  // (B,100)
  const int*   boxes   = (const int*)  d_in[2];  // (B,100,4)
  const float* masks   = (const float*)d_in[3];  // (B,100,28,28)
  const float* seg     = (const float*)d_in[4];  // (B,160,160,20)
  const int B = in_sizes[0] / NDET;

  Det*      dets = (Det*)d_ws;                                     // B*100*48 B
  unsigned* bits = (unsigned*)((char*)d_ws + (size_t)B * NDET * sizeof(Det)); // B*100*28*4 B (16B-aligned: 9600 % 16 == 0)

  panoptic_prep<<<dim3(B), dim3(256), 0, stream>>>(scores, classes, boxes, masks,
                                                   dets, bits);

  dim3 grid(OUT_HW / 16, OUT_HW / 16, B);
  panoptic_main<<<grid, dim3(256), 0, stream>>>(seg, dets, bits,
                                                (int*)d_out, B);
}